// HGraphSAGE_64415919506091
// MI455X (gfx1250) — compile-verified
//
#include <hip/hip_runtime.h>
#include <math.h>

#define N_SRCN 100000
#define N_DSTN 100000
#define NE     1000000
#define DD     128
#define RR     3
#define NEG_SLOPE 0.2f

typedef __attribute__((ext_vector_type(2))) float v2f;
typedef __attribute__((ext_vector_type(8))) float v8f;

// ---------- float atomic max via monotonic int mapping ----------
__device__ inline void atomicMaxF(float* addr, float v) {
  if (v >= 0.0f) {
    atomicMax((int*)addr, __float_as_int(v));
  } else {
    atomicMin((unsigned int*)addr, __float_as_uint(v));
  }
}

// ---------- init: zero rst, -inf segmax, zero segsum/wsum ----------
__global__ void init_ws_k(float* __restrict__ z, float* __restrict__ m,
                          float* __restrict__ s, float* __restrict__ wsum) {
  long long idx = (long long)blockIdx.x * blockDim.x + threadIdx.x;
  long long nz = (long long)RR * N_DSTN * DD;
  if (idx < nz) z[idx] = 0.0f;
  if (idx < (long long)RR * N_DSTN) { m[idx] = -__builtin_inff(); s[idx] = 0.0f; }
  if (idx < RR) wsum[idx] = 0.0f;
}

// ---------- wal[r][k] = sum_j W[r][k][j]*al[r][j] ; same for war ----------
__global__ void wal_war_k(const float* __restrict__ W, const float* __restrict__ al,
                          const float* __restrict__ ar, float* __restrict__ wal,
                          float* __restrict__ war) {
  int r = blockIdx.x;
  int k = threadIdx.x;
  const float* Wr = W + ((long long)r * DD + k) * DD;
  float sl = 0.0f, sr = 0.0f;
  for (int j = 0; j < DD; ++j) {
    float w = Wr[j];
    sl += w * al[r * DD + j];
    sr += w * ar[r * DD + j];
  }
  wal[r * DD + k] = sl;
  war[r * DD + k] = sr;
}

// ---------- FP32 WMMA GEMM: C[r] = A[r] @ W[r] (+ bias per column) ----------
// A: [nrows,128] row-major, W: [128,128] row-major, C: [nrows,128].
// nrows must be a multiple of 16 (100000 = 6250*16).
__global__ __launch_bounds__(256) void gemm_wmma_k(
    const float* __restrict__ A, long long relA,
    const float* __restrict__ W, long long relW,
    const float* __restrict__ bias,
    float* __restrict__ C, long long relC, int nrows) {
  // LDS holds W pre-swizzled into B-fragment order:
  // frag(kt,nt): lane l<16 -> (K=4kt+{0,1}, N=nt*16+l); l>=16 -> (K=4kt+{2,3}, N=nt*16+l-16)
  __shared__ float lw[16384];
  const int r = blockIdx.y;
  const float* Wr = W + (long long)r * relW;
  for (int idx = threadIdx.x; idx < 16384; idx += 256) {
    int i    = idx & 1;
    int lane = (idx >> 1) & 31;
    int nt   = (idx >> 6) & 7;
    int kt   = idx >> 9;
    int krow = 4 * kt + ((lane >= 16) ? 2 : 0) + i;
    int col  = nt * 16 + (lane & 15);
    lw[idx] = Wr[krow * DD + col];
  }
  __syncthreads();

  const int wave = threadIdx.x >> 5;
  const int lane = threadIdx.x & 31;
  long long strip = (long long)blockIdx.x * 8 + wave;   // 16-row strip (wave-uniform)
  if (strip * 16 >= nrows) return;                       // uniform branch: EXEC all-1 for WMMA

  const float* Ar = A + (long long)r * relA + strip * 16 * DD;
  const int aRow = lane & 15;
  const int aK   = (lane >= 16) ? 2 : 0;

  v8f acc[8] = {};
  for (int kt = 0; kt < 32; ++kt) {
    const float* ap = Ar + aRow * DD + kt * 4 + aK;      // 8B aligned
    v2f af;
    af.x = ap[0];
    af.y = ap[1];
    const float* bp = &lw[(kt * 8) * 64 + lane * 2];
#pragma unroll
    for (int nt = 0; nt < 8; ++nt) {
      v2f bf;
      bf.x = bp[nt * 64];
      bf.y = bp[nt * 64 + 1];
      acc[nt] = __builtin_amdgcn_wmma_f32_16x16x4_f32(
          false, af, false, bf, (short)0, acc[nt], false, false);
    }
  }

  float* Cr = C + (long long)r * relC + strip * 16 * DD;
  const int colL   = lane & 15;
  const int rowOff = (lane >= 16) ? 8 : 0;
#pragma unroll
  for (int nt = 0; nt < 8; ++nt) {
    int col = nt * 16 + colL;
    float b = bias ? bias[col] : 0.0f;
#pragma unroll
    for (int m = 0; m < 8; ++m)
      Cr[(rowOff + m) * DD + col] = acc[nt][m] + b;
  }
}

// ---------- out[r][i] = dot(X[r][i], v[r]) (wave per row) ----------
__global__ __launch_bounds__(256) void rowdot_k(const float* __restrict__ X, long long relX,
                                                const float* __restrict__ v,
                                                float* __restrict__ out, int n) {
  int r = blockIdx.y;
  int wave = threadIdx.x >> 5, lane = threadIdx.x & 31;
  long long node = (long long)blockIdx.x * 8 + wave;
  if (node >= n) return;
  const float4 x = *(const float4*)(X + (long long)r * relX + node * DD + lane * 4);
  const float4 w = *(const float4*)(v + r * DD + lane * 4);
  float s = x.x * w.x + x.y * w.y + x.z * w.z + x.w * w.w;
  for (int off = 16; off; off >>= 1) s += __shfl_down(s, off);
  if (lane == 0) out[(long long)r * n + node] = s;
}

// ---------- edge pass 1: e = leaky(el[s]+er[d]); segmax ----------
__global__ void edge_pass1_k(const int* __restrict__ es, const int* __restrict__ ed,
                             const float* __restrict__ el, const float* __restrict__ er,
                             float* __restrict__ ebuf, float* __restrict__ mmax) {
  int r = blockIdx.y;
  long long t = (long long)blockIdx.x * blockDim.x + threadIdx.x;
  if (t >= NE) return;
  int s = es[(long long)r * NE + t];
  int d = ed[(long long)r * NE + t];
  float v = el[(long long)r * N_SRCN + s] + er[(long long)r * N_DSTN + d];
  v = (v > 0.0f) ? v : NEG_SLOPE * v;
  ebuf[(long long)r * NE + t] = v;
  atomicMaxF(&mmax[(long long)r * N_DSTN + d], v);
}

// ---------- edge pass 2: ex = exp(e - m[d]); segsum ----------
__global__ void edge_pass2_k(const int* __restrict__ ed, float* __restrict__ ebuf,
                             const float* __restrict__ mmax, float* __restrict__ ssum) {
  int r = blockIdx.y;
  long long t = (long long)blockIdx.x * blockDim.x + threadIdx.x;
  if (t >= NE) return;
  int d = ed[(long long)r * NE + t];
  float ex = __expf(ebuf[(long long)r * NE + t] - mmax[(long long)r * N_DSTN + d]);
  ebuf[(long long)r * NE + t] = ex;
  atomicAdd(&ssum[(long long)r * N_DSTN + d], ex);
}

// ---------- edge pass 3: rst[d] += alpha * h_src[s] (32 lanes per edge) ----------
__global__ __launch_bounds__(256) void edge_pass3_k(
    const int* __restrict__ es, const int* __restrict__ ed,
    const float* __restrict__ ebuf, const float* __restrict__ ssum,
    const float* __restrict__ hsrc, float* __restrict__ rst) {
  int r = blockIdx.y;
  long long t = (long long)blockIdx.x * 256 + threadIdx.x;
  if (t >= (long long)NE * 32) return;
  int lane = (int)(t & 31);
  long long e = t >> 5;
  int s = es[(long long)r * NE + e];
  int d = ed[(long long)r * NE + e];
  float alpha = ebuf[(long long)r * NE + e] / ssum[(long long)r * N_DSTN + d];
  const float4 h = *(const float4*)(hsrc + ((long long)r * N_SRCN + s) * DD + lane * 4);
  float* dst = rst + ((long long)r * N_DSTN + d) * DD + lane * 4;
  atomicAdd(dst + 0, alpha * h.x);
  atomicAdd(dst + 1, alpha * h.y);
  atomicAdd(dst + 2, alpha * h.z);
  atomicAdd(dst + 3, alpha * h.w);
}

// ---------- z = elu(rst + gat_bias) in place ----------
__global__ void finalize_z_k(float* __restrict__ z, const float* __restrict__ bias) {
  long long idx = (long long)blockIdx.x * blockDim.x + threadIdx.x;
  long long nz = (long long)RR * N_DSTN * DD;
  if (idx >= nz) return;
  int c = (int)(idx % DD);
  int r = (int)(idx / ((long long)N_DSTN * DD));
  float v = z[idx] + bias[r * DD + c];
  z[idx] = (v > 0.0f) ? v : expm1f(v);
}

// ---------- semantic reduce: wsum[r] += sum_n tanh(T[r][n]) . w2 ----------
__global__ __launch_bounds__(256) void sem_reduce_k(const float* __restrict__ T,
                                                    const float* __restrict__ w2,
                                                    float* __restrict__ wsum, int n) {
  int r = blockIdx.y;
  int wave = threadIdx.x >> 5, lane = threadIdx.x & 31;
  __shared__ float red[8];
  float acc = 0.0f;
  long long base = (long long)blockIdx.x * 128;
  for (int i = 0; i < 16; ++i) {
    long long node = base + wave * 16 + i;
    if (node < n) {
      const float4 t = *(const float4*)(T + ((long long)r * n + node) * DD + lane * 4);
      const float4 w = *(const float4*)(w2 + lane * 4);
      acc += tanhf(t.x) * w.x + tanhf(t.y) * w.y + tanhf(t.z) * w.z + tanhf(t.w) * w.w;
    }
  }
  for (int off = 16; off; off >>= 1) acc += __shfl_down(acc, off);
  if (lane == 0) red[wave] = acc;
  __syncthreads();
  if (threadIdx.x == 0) {
    float s = 0.0f;
    for (int i = 0; i < 8; ++i) s += red[i];
    atomicAdd(&wsum[r], s);
  }
}

// ---------- softmax over R; also emit att_mp ----------
__global__ void sem_softmax_k(const float* __restrict__ wsum, float* __restrict__ a,
                              float* __restrict__ out_att) {
  if (threadIdx.x == 0) {
    float w[RR], m = -__builtin_inff();
    for (int i = 0; i < RR; ++i) { w[i] = wsum[i] / (float)N_DSTN; m = fmaxf(m, w[i]); }
    float s = 0.0f, e[RR];
    for (int i = 0; i < RR; ++i) { e[i] = __expf(w[i] - m); s += e[i]; }
    for (int i = 0; i < RR; ++i) { a[i] = e[i] / s; out_att[i] = e[i] / s; }
  }
}

// ---------- z_out = sum_r a[r]*z[r] ----------
__global__ void combine_k(const float* __restrict__ z, const float* __restrict__ a,
                          float* __restrict__ out) {
  long long idx = (long long)blockIdx.x * blockDim.x + threadIdx.x;
  long long nd = (long long)N_DSTN * DD;
  if (idx >= nd) return;
  out[idx] = a[0] * z[idx] + a[1] * z[nd + idx] + a[2] * z[2 * nd + idx];
}

extern "C" void kernel_launch(void* const* d_in, const int* in_sizes, int n_in,
                              void* d_out, int out_size, void* d_ws, size_t ws_size,
                              hipStream_t stream) {
  const float* dst_feat = (const float*)d_in[0];
  const float* src_feat = (const float*)d_in[1];
  const float* fc_W     = (const float*)d_in[2];
  const float* attn_l   = (const float*)d_in[3];
  const float* attn_r   = (const float*)d_in[4];
  const float* gat_bias = (const float*)d_in[5];
  const float* sem_W1   = (const float*)d_in[6];
  const float* sem_b1   = (const float*)d_in[7];
  const float* sem_w2   = (const float*)d_in[8];
  const int*   edge_src = (const int*)d_in[9];
  const int*   edge_dst = (const int*)d_in[10];
  float* out = (float*)d_out;
  (void)in_sizes; (void)n_in; (void)out_size; (void)ws_size;

  char* ws = (char*)d_ws;
  size_t off = 0;
  auto alloc = [&](size_t bytes) -> char* {
    char* p = ws + off;
    off = (off + bytes + 255) & ~(size_t)255;
    return p;
  };
  float* h_src = (float*)alloc((size_t)RR * N_SRCN * DD * 4);  // later reused as Tsem
  float* z     = (float*)alloc((size_t)RR * N_DSTN * DD * 4);
  float* el    = (float*)alloc((size_t)RR * N_SRCN * 4);
  float* er    = (float*)alloc((size_t)RR * N_DSTN * 4);
  float* mmax  = (float*)alloc((size_t)RR * N_DSTN * 4);
  float* ssum  = (float*)alloc((size_t)RR * N_DSTN * 4);
  float* ebuf  = (float*)alloc((size_t)RR * NE * 4);
  float* wal   = (float*)alloc((size_t)RR * DD * 4);
  float* war   = (float*)alloc((size_t)RR * DD * 4);
  float* wsum  = (float*)alloc((size_t)RR * 4);
  float* aw    = (float*)alloc((size_t)RR * 4);

  long long nz = (long long)RR * N_DSTN * DD;

  init_ws_k<<<dim3((unsigned)((nz + 255) / 256)), 256, 0, stream>>>(z, mmax, ssum, wsum);
  wal_war_k<<<dim3(RR), dim3(DD), 0, stream>>>(fc_W, attn_l, attn_r, wal, war);

  // h_src[r] = src_feat[r] @ fc_W[r]   (FP32 WMMA)
  gemm_wmma_k<<<dim3((N_SRCN / 16 + 7) / 8, RR), 256, 0, stream>>>(
      src_feat, (long long)N_SRCN * DD, fc_W, (long long)DD * DD, nullptr,
      h_src, (long long)N_SRCN * DD, N_SRCN);

  rowdot_k<<<dim3((N_SRCN + 7) / 8, RR), 256, 0, stream>>>(
      src_feat, (long long)N_SRCN * DD, wal, el, N_SRCN);
  rowdot_k<<<dim3((N_DSTN + 7) / 8, RR), 256, 0, stream>>>(
      dst_feat, 0, war, er, N_DSTN);

  edge_pass1_k<<<dim3((NE + 255) / 256, RR), 256, 0, stream>>>(edge_src, edge_dst, el, er, ebuf, mmax);
  edge_pass2_k<<<dim3((NE + 255) / 256, RR), 256, 0, stream>>>(edge_dst, ebuf, mmax, ssum);
  edge_pass3_k<<<dim3((unsigned)(((long long)NE * 32 + 255) / 256), RR), 256, 0, stream>>>(
      edge_src, edge_dst, ebuf, ssum, h_src, z);

  finalize_z_k<<<dim3((unsigned)((nz + 255) / 256)), 256, 0, stream>>>(z, gat_bias);

  // Tsem[r] = z[r] @ sem_W1 + sem_b1   (FP32 WMMA, h_src buffer recycled)
  float* Tsem = h_src;
  gemm_wmma_k<<<dim3((N_DSTN / 16 + 7) / 8, RR), 256, 0, stream>>>(
      z, (long long)N_DSTN * DD, sem_W1, 0, sem_b1,
      Tsem, (long long)N_DSTN * DD, N_DSTN);

  sem_reduce_k<<<dim3((N_DSTN + 127) / 128, RR), 256, 0, stream>>>(Tsem, sem_w2, wsum, N_DSTN);
  sem_softmax_k<<<1, 32, 0, stream>>>(wsum, aw, out + (long long)N_DSTN * DD);
  combine_k<<<dim3((unsigned)(((long long)N_DSTN * DD + 255) / 256)), 256, 0, stream>>>(z, aw, out);
}